// MoELayer_78460462564083
// MI455X (gfx1250) — compile-verified
//
#include <hip/hip_runtime.h>
#include <math.h>

// Problem constants (match reference)
#define BTOK 8192
#define DD   2048
#define HH   2048
#define EE   8

// GEMM tiling
#define BM 128
#define BN 128
#define BK 32
#define LDT 36   // padded LDS row stride in floats (144B: 16B-aligned, bank-spread)

typedef float v2f __attribute__((ext_vector_type(2)));
typedef float v8f __attribute__((ext_vector_type(8)));

// ---------------------------------------------------------------------------
// CDNA5 async global->LDS copy (16B per lane), tracked by ASYNCcnt.
// Inline asm keeps it portable across ROCm 7.2 / amdgpu-toolchain builtins.
// ---------------------------------------------------------------------------
__device__ __forceinline__ void async_g2l_b128(unsigned lds_byte_addr, const float* gptr) {
    asm volatile("global_load_async_to_lds_b128 %0, %1, off"
                 :: "v"(lds_byte_addr), "v"(gptr)
                 : "memory");
}
__device__ __forceinline__ void wait_asynccnt0() {
#if __has_builtin(__builtin_amdgcn_s_wait_asynccnt)
    __builtin_amdgcn_s_wait_asynccnt(0);
#else
    asm volatile("s_wait_asynccnt 0x0" ::: "memory");
#endif
}
__device__ __forceinline__ unsigned lds_addr_of(const void* p) {
    // Generic pointers to LDS carry the LDS byte offset in their low 32 bits
    // (flat aperture rule).
    return (unsigned)(unsigned long long)p;
}

// ---------------------------------------------------------------------------
// Kernel 0: zero output (we accumulate with atomics) and per-expert counters
// ---------------------------------------------------------------------------
__global__ void moe_zero_kernel(float* __restrict__ out, int* __restrict__ counts) {
    size_t i = (size_t)blockIdx.x * blockDim.x + threadIdx.x;
    const size_t n4 = (size_t)BTOK * HH / 4;
    float4 z; z.x = 0.f; z.y = 0.f; z.z = 0.f; z.w = 0.f;
    for (size_t p = i; p < n4; p += (size_t)gridDim.x * blockDim.x)
        ((float4*)out)[p] = z;
    if (blockIdx.x == 0 && threadIdx.x < EE) counts[threadIdx.x] = 0;
}

// ---------------------------------------------------------------------------
// Kernel 1: gating. One wave32 per token: 8 dot products over D, cross-lane
// reduce, stable top-2, softmax over the 2 logits, append to expert lists.
// ---------------------------------------------------------------------------
__global__ __launch_bounds__(256)
void moe_gate_kernel(const float* __restrict__ x,
                     const float* __restrict__ gate_W,
                     const float* __restrict__ gate_b,
                     int* __restrict__ counts,
                     int* __restrict__ lists,
                     float* __restrict__ wts) {
    const int wave = threadIdx.x >> 5;
    const int lane = threadIdx.x & 31;
    const int tok  = blockIdx.x * 8 + wave;
    const float* xr = x + (size_t)tok * DD;

    float acc[EE];
#pragma unroll
    for (int e = 0; e < EE; ++e) acc[e] = 0.f;

    for (int d = lane; d < DD; d += 32) {
        const float xv = xr[d];
#pragma unroll
        for (int e = 0; e < EE; ++e)
            acc[e] = fmaf(xv, gate_W[e * DD + d], acc[e]);
    }
#pragma unroll
    for (int e = 0; e < EE; ++e) {
#pragma unroll
        for (int m = 16; m >= 1; m >>= 1)
            acc[e] += __shfl_xor(acc[e], m, 32);
    }

    if (lane == 0) {
        float v0 = -1e30f, v1 = -1e30f;
        int   i0 = 0,      i1 = 0;
#pragma unroll
        for (int e = 0; e < EE; ++e) {
            const float v = acc[e] + gate_b[e];
            if (v > v0)      { v1 = v0; i1 = i0; v0 = v; i0 = e; }
            else if (v > v1) { v1 = v;  i1 = e; }
        }
        const float e1  = __expf(v1 - v0);     // v0 is the max
        const float inv = 1.f / (1.f + e1);
        const float w0 = inv;
        const float w1 = e1 * inv;

        int p0 = atomicAdd(&counts[i0], 1);
        lists[i0 * BTOK + p0] = tok;  wts[i0 * BTOK + p0] = w0;
        int p1 = atomicAdd(&counts[i1], 1);
        lists[i1 * BTOK + p1] = tok;  wts[i1 * BTOK + p1] = w1;
    }
}

// ---------------------------------------------------------------------------
// Kernel 2: per-expert gathered GEMM via V_WMMA_F32_16X16X4_F32 with
// double-buffered LDS fed by GLOBAL_LOAD_ASYNC_TO_LDS_B128 (ASYNCcnt).
// Block = 256 threads = 8 waves; block tile 128(tok) x 128(H) over full D.
// Wave (wm,wn) owns a 32x64 sub-tile = 2x4 accumulators of 16x16.
// Epilogue: out[tok] += w * (acc + bias_e) via global f32 atomic adds.
// ---------------------------------------------------------------------------
__global__ __launch_bounds__(256)
void moe_expert_kernel(const float* __restrict__ x,
                       const float* __restrict__ expert_W,
                       const float* __restrict__ expert_b,
                       const int* __restrict__ counts,
                       const int* __restrict__ lists,
                       const float* __restrict__ wts,
                       float* __restrict__ out) {
    const int e   = blockIdx.z;
    const int cnt = counts[e];
    const int m0  = blockIdx.x * BM;
    if (m0 >= cnt) return;                      // block-uniform early exit
    const int n0  = blockIdx.y * BN;

    __shared__ float sA[2][BM * LDT];
    __shared__ float sB[2][BN * LDT];
    __shared__ int   sTok[BM];
    __shared__ float sW[BM];

    const int tid = threadIdx.x;

    if (tid < BM) {
        const int r = m0 + tid;
        int t = 0; float w = 0.f;
        if (r < cnt) { t = lists[e * BTOK + r]; w = wts[e * BTOK + r]; }
        sTok[tid] = t; sW[tid] = w;             // padded rows: tok 0, weight 0
    }
    __syncthreads();

    const float* Wbase = expert_W + (size_t)e * HH * DD;

    const int wave   = tid >> 5;
    const int lane   = tid & 31;
    const int wm     = wave & 3;                // 4 row groups of 32
    const int wn     = wave >> 2;               // 2 col groups of 64
    const int laneLo = lane & 15;
    const int kHalf  = (lane >> 4) << 1;        // lanes 16-31 take K+2,K+3

    // Per-thread stage-copy geometry: 4 chunks of 16B for A, 4 for B.
    const int rowA = tid >> 3;                  // two rows per 16-thread group... (id>>3 below)
    (void)rowA;

    v8f acc[2][4];
#pragma unroll
    for (int i = 0; i < 2; ++i)
#pragma unroll
        for (int j = 0; j < 4; ++j)
#pragma unroll
            for (int q = 0; q < 8; ++q) acc[i][j][q] = 0.f;

    // Issue one stage of async copies into buffer `buf` for K-offset k0.
    auto issue_stage = [&](int buf, int k0) {
        const unsigned aBase = lds_addr_of(&sA[buf][0]);
        const unsigned bBase = lds_addr_of(&sB[buf][0]);
#pragma unroll
        for (int p = 0; p < 4; ++p) {
            const int id  = tid + p * 256;      // 1024 16B chunks: 128 rows x 8
            const int row = id >> 3;
            const int c   = (id & 7) << 2;
            async_g2l_b128(aBase + (unsigned)(row * LDT + c) * 4u,
                           x + (size_t)sTok[row] * DD + k0 + c);
        }
#pragma unroll
        for (int p = 0; p < 4; ++p) {
            const int id  = tid + p * 256;
            const int col = id >> 3;
            const int c   = (id & 7) << 2;
            async_g2l_b128(bBase + (unsigned)(col * LDT + c) * 4u,
                           Wbase + (size_t)(n0 + col) * DD + k0 + c);
        }
    };

    issue_stage(0, 0);                          // prologue
    int cur = 0;

    for (int k0 = 0; k0 < DD; k0 += BK) {
        wait_asynccnt0();                       // buffer `cur` data landed
        __syncthreads();                        // ... for every wave
        if (k0 + BK < DD) issue_stage(cur ^ 1, k0 + BK);

        const float* aBuf = sA[cur];
        const float* bBuf = sB[cur];
#pragma unroll
        for (int k = 0; k < BK; k += 4) {
            v2f a[2], b[4];
#pragma unroll
            for (int i = 0; i < 2; ++i)
                a[i] = *(const v2f*)&aBuf[(wm * 32 + i * 16 + laneLo) * LDT + k + kHalf];
#pragma unroll
            for (int j = 0; j < 4; ++j)
                b[j] = *(const v2f*)&bBuf[(wn * 64 + j * 16 + laneLo) * LDT + k + kHalf];
#pragma unroll
            for (int i = 0; i < 2; ++i)
#pragma unroll
                for (int j = 0; j < 4; ++j)
                    acc[i][j] = __builtin_amdgcn_wmma_f32_16x16x4_f32(
                        false, a[i], false, b[j], (short)0, acc[i][j], false, false);
        }
        __syncthreads();                        // all waves done with `cur`
        cur ^= 1;
    }

    // Epilogue: C/D layout — vgpr r, lanes 0-15: (M=r, N=lane); lanes 16-31: (M=8+r, N=lane-16)
    const int rowHalf = (lane >> 4) * 8;
#pragma unroll
    for (int i = 0; i < 2; ++i) {
#pragma unroll
        for (int j = 0; j < 4; ++j) {
            const int gcol = n0 + wn * 64 + j * 16 + laneLo;
            const float bias = expert_b[e * HH + gcol];
#pragma unroll
            for (int r = 0; r < 8; ++r) {
                const int rowL = wm * 32 + i * 16 + rowHalf + r;
                const int tok  = sTok[rowL];
                const float w  = sW[rowL];
                const float val = w * (acc[i][j][r] + bias);
                unsafeAtomicAdd(&out[(size_t)tok * HH + gcol], val);
            }
        }
    }
}

// ---------------------------------------------------------------------------
extern "C" void kernel_launch(void* const* d_in, const int* in_sizes, int n_in,
                              void* d_out, int out_size, void* d_ws, size_t ws_size,
                              hipStream_t stream) {
    const float* x        = (const float*)d_in[0];
    const float* expert_W = (const float*)d_in[1];
    const float* expert_b = (const float*)d_in[2];
    const float* gate_W   = (const float*)d_in[3];
    const float* gate_b   = (const float*)d_in[4];
    float* out = (float*)d_out;

    char* ws      = (char*)d_ws;
    int*   counts = (int*)ws;                                      // 8 ints
    int*   lists  = (int*)(ws + 1024);                             // E*B ints
    float* wts    = (float*)(ws + 1024 + sizeof(int) * EE * BTOK); // E*B floats

    moe_zero_kernel<<<2048, 256, 0, stream>>>(out, counts);
    moe_gate_kernel<<<BTOK / 8, 256, 0, stream>>>(x, gate_W, gate_b, counts, lists, wts);

    dim3 grid(BTOK / BM, HH / BN, EE);
    moe_expert_kernel<<<grid, 256, 0, stream>>>(x, expert_W, expert_b,
                                                counts, lists, wts, out);
}